// PitchforkBifurcation_82059645157925
// MI455X (gfx1250) — compile-verified
//
#include <hip/hip_runtime.h>
#include <hip/hip_bf16.h>
#include <math.h>
#include <stdint.h>

// Pitchfork bifurcation + Poincare expmap, bandwidth-optimal single pass over h.
// - Tile staging:   one TDM tensor_load_to_lds (iterate mode) per 16x512 tile,
//                   padding rows to 516 floats in LDS (bank-conflict-free).
// - Reductions:     V_WMMA_F32_16X16X4_F32 with all-ones B (layout-robust row sums)
//                   giving x = h.w and q = ||h||^2 per row in one pass.
// - Scalars:        RK4 + clip + expmap coefficients; ||h'||^2 = q + 2dx + d^2||w||^2.
// - Output:         elementwise coef*(h + delta*w), non-temporal b128 stores.

typedef __attribute__((ext_vector_type(2))) float v2f;
typedef __attribute__((ext_vector_type(4))) float v4f;
typedef __attribute__((ext_vector_type(8))) float v8f;
typedef __attribute__((ext_vector_type(4))) unsigned int v4u;
typedef __attribute__((ext_vector_type(4))) int v4i;
typedef __attribute__((ext_vector_type(8))) int v8i;

#define TILE_ROWS 16
#define DIM       512
#define LDS_P     516   // padded row stride in floats (multiple of 4, stride-4 banks)
#define WAVES     4
#define BLOCK     128

#if __has_include(<hip/amd_detail/amd_gfx1250_TDM.h>)
#define TDM_6ARG 1
#endif

// One TDM descriptor: 16 iterations of a 1x512 f32 row; global rows are
// contiguous (512 floats), LDS rows padded to 516 floats.
__device__ __forceinline__ void tdm_load_tile(const float* gsrc, unsigned int lds_off)
{
    const unsigned long long ga = (unsigned long long)(uintptr_t)gsrc;
    v4u g0;
    g0.x = 1u;                                    // count=1 (valid user descriptor)
    g0.y = lds_off;                               // lds_addr (bytes)
    g0.z = (unsigned int)(ga & 0xFFFFFFFFu);      // global_addr[31:0]
    g0.w = (unsigned int)((ga >> 32) & 0x01FFFFFFu)  // global_addr[56:32]
         | 0x80000000u;                           // type=2 ("image")

    v8i g1;
    g1[0] = (2 << 16) | (1 << 19);                // data_size=4B, iterate_enable=1
    g1[1] = (int)(512u << 16);                    // tensor_dim0 = 512 (bits[63:48]->[31:16])
    g1[2] = (int)(1u << 16);                      // tensor_dim0 hi=0 | tensor_dim1 = 1
    g1[3] = (int)(512u << 16);                    // tensor_dim1 hi=0 | tile_dim0 = 512
    g1[4] = 1;                                    // tile_dim1 = 1, tile_dim2 = 0
    g1[5] = 512;                                  // tensor_dim0_stride = 512 elems
    g1[6] = 0;                                    // stride hi | tensor_dim1_stride lo
    g1[7] = 0;                                    // tensor_dim1_stride hi

    v4i g2;
    g2[0] = 1;                                    // tensor_dim2 (unused)
    g2[1] = LDS_P;                                // lds_addr_increment = 516 elems
    g2[2] = DIM;                                  // global_addr_increment = 512 elems
    g2[3] = (int)((unsigned)(TILE_ROWS - 1) << 16); // iterate_count = 15 -> 16 iters

    v4i g3; g3[0] = 0; g3[1] = 0; g3[2] = 0; g3[3] = 0;

#ifdef TDM_6ARG
    v8i g4 = {};
    __builtin_amdgcn_tensor_load_to_lds(g0, g1, g2, g3, g4, 0);
#else
    __builtin_amdgcn_tensor_load_to_lds(g0, g1, g2, g3, 0);
#endif
}

__global__ __launch_bounds__(BLOCK)
void PitchforkBifurcation_82059645157925_kernel(
    const float* __restrict__ h,        // [R, 512]
    const float* __restrict__ tension,  // [R]
    const float* __restrict__ w,        // [512]
    const float* __restrict__ tscale,   // [1]
    const float* __restrict__ noise,    // [R]
    float* __restrict__ out)            // [R, 512]
{
    __shared__ float sh_h[TILE_ROWS * LDS_P];   // 33,024 B (padded tile)
    __shared__ float sh_w[DIM];                 //  2,048 B
    __shared__ float sp_x[WAVES][TILE_ROWS];
    __shared__ float sp_q[WAVES][TILE_ROWS];
    __shared__ float sp_w2[WAVES * 2];
    __shared__ float s_delta[TILE_ROWS];
    __shared__ float s_coef[TILE_ROWS];

    const int tid  = threadIdx.x;        // 0..127
    const int lane = tid & 31;
    const int wave = tid >> 5;
    const long row0 = (long)blockIdx.x * TILE_ROWS;

    // ---- TDM: stage the whole 16x512 tile into padded LDS (wave 0 only) ----
    if (wave == 0) {
        tdm_load_tile(h + row0 * DIM,
                      (unsigned int)(uintptr_t)(const void*)sh_h);
    }
    // ---- stage w into LDS (128 lanes x float4 = 512 floats) ----
    {
        v4f wv = ((const v4f*)w)[tid];
        ((v4f*)sh_w)[tid] = wv;
    }
    if (wave == 0) {
        __builtin_amdgcn_s_wait_tensorcnt(0);    // tile resident in LDS
    }
    __syncthreads();                             // other waves wait on wave 0

    // ---- WMMA reduction pass: each wave owns a 128-column quarter ----
    const int half = lane >> 4;          // 0/1 -> K pair (0,1) vs (2,3)
    const int m    = lane & 15;          // A-matrix row for this lane

    v8f acc_x = {};                      // D[m][n] += sum_k h[m][k]*w[k]
    v8f acc_q = {};                      // D[m][n] += sum_k h[m][k]^2
    v2f ones; ones.x = 1.0f; ones.y = 1.0f;
    float wp2 = 0.0f;

    const float* aptr = sh_h + m * LDS_P + wave * 128 + 2 * half;
    const float* wptr = sh_w + wave * 128 + 2 * half;

    #pragma unroll 8
    for (int s = 0; s < 32; ++s) {
        v2f a  = *(const v2f*)aptr;      // A frag: lanes0-15 K=0,1 ; lanes16-31 K=2,3
        v2f wv = *(const v2f*)wptr;      // matching w pair
        v2f aw; aw.x = a.x * wv.x; aw.y = a.y * wv.y;
        v2f a2; a2.x = a.x * a.x;  a2.y = a.y * a.y;
        wp2 += wv.x * wv.x + wv.y * wv.y;
        acc_x = __builtin_amdgcn_wmma_f32_16x16x4_f32(
            false, aw, false, ones, (short)0, acc_x, false, false);
        acc_q = __builtin_amdgcn_wmma_f32_16x16x4_f32(
            false, a2, false, ones, (short)0, acc_q, false, false);
        aptr += 4; wptr += 4;
    }

    // Column N=0 of D: lane 0 holds rows 0-7 (vgpr 0..7), lane 16 holds rows 8-15.
    if (m == 0) {
        #pragma unroll
        for (int j = 0; j < 8; ++j) {
            sp_x[wave][half * 8 + j] = acc_x[j];
            sp_q[wave][half * 8 + j] = acc_q[j];
        }
        sp_w2[wave * 2 + half] = wp2;
    }
    __syncthreads();

    // ---- per-row scalar pipeline (RK4 + clip + expmap coefficients) ----
    if (tid < TILE_ROWS) {
        float x = sp_x[0][tid] + sp_x[1][tid] + sp_x[2][tid] + sp_x[3][tid];
        float q = sp_q[0][tid] + sp_q[1][tid] + sp_q[2][tid] + sp_q[3][tid];
        float W2 = sp_w2[0] + sp_w2[1] + sp_w2[2] + sp_w2[3]
                 + sp_w2[4] + sp_w2[5] + sp_w2[6] + sp_w2[7];
        const long row = row0 + tid;
        const float tn = tension[row];
        const float nz = noise[row];
        const float ts = tscale[0];

        float r  = fminf(fmaxf(tn, -10.0f), 2.0f);
        float xi = (x == 0.0f) ? nz * 1e-4f : x;
        #pragma unroll
        for (int it = 0; it < 4; ++it) {           // RK4, f(z)=0.5*(r z - z^3)
            float k1 = 0.5f * (r * xi - xi * xi * xi);
            float z2 = xi + 0.5f * k1;
            float k2 = 0.5f * (r * z2 - z2 * z2 * z2);
            float z3 = xi + 0.5f * k2;
            float k3 = 0.5f * (r * z3 - z3 * z3 * z3);
            float z4 = xi + k3;
            float k4 = 0.5f * (r * z4 - z4 * z4 * z4);
            xi = xi + (k1 + 2.0f * k2 + 2.0f * k3 + k4) * (1.0f / 6.0f);
        }
        float delta = xi - x;
        // ||h + delta*w||^2 = q + 2*delta*x + delta^2*||w||^2
        float hp2 = fmaxf(q + 2.0f * delta * x + delta * delta * W2, 0.0f);
        float ng   = fabsf(ts) * sqrtf(hp2);               // ||h' * tscale||
        float sc   = fminf(4.0f / (ng + 1e-8f), 1.0f);     // tangent clip
        float nv   = ng * sc;                              // ||clipped||
        float safe = fmaxf(nv, 1e-15f);
        float coef = tanhf(safe) / safe * sc * ts;         // expmap0 scale
        s_delta[tid] = delta;
        s_coef[tid]  = coef;
    }
    __syncthreads();

    // ---- elementwise output: out = coef * (h + delta*w), NT b128 streaming ----
    #pragma unroll
    for (int rr = 0; rr < TILE_ROWS / WAVES; ++rr) {
        const int r  = wave * (TILE_ROWS / WAVES) + rr;
        const float dl = s_delta[r];
        const float cf = s_coef[r];
        const v4f* hr = (const v4f*)(sh_h + r * LDS_P);
        const v4f* wr = (const v4f*)sh_w;
        v4f* og = (v4f*)(out + (row0 + r) * DIM);
        #pragma unroll
        for (int c = lane; c < DIM / 4; c += 32) {
            v4f hv = hr[c];
            v4f wv = wr[c];
            v4f o  = cf * (hv + dl * wv);
            __builtin_nontemporal_store(o, &og[c]);
        }
    }
}

extern "C" void kernel_launch(void* const* d_in, const int* in_sizes, int n_in,
                              void* d_out, int out_size, void* d_ws, size_t ws_size,
                              hipStream_t stream) {
    const float* h   = (const float*)d_in[0];   // h_comp  [B,S,512]
    const float* tn  = (const float*)d_in[1];   // tension [B,S,1]
    const float* w   = (const float*)d_in[2];   // w_proj  [512]
    const float* ts  = (const float*)d_in[3];   // tangent_scale scalar
    const float* nz  = (const float*)d_in[4];   // noise   [B,S,1]
    float* out = (float*)d_out;

    const int nrows = in_sizes[1];              // B*S
    const int tiles = nrows / TILE_ROWS;        // 4096 for the reference shapes

    hipLaunchKernelGGL(PitchforkBifurcation_82059645157925_kernel,
                       dim3(tiles), dim3(BLOCK), 0, stream,
                       h, tn, w, ts, nz, out);
}